// MultiHeadAttentionQuantum_65481071400570
// MI455X (gfx1250) — compile-verified
//
#include <hip/hip_runtime.h>
#include <math.h>

// MultiHeadAttentionQuantum for MI455X (gfx1250, wave32).
// Pipeline:
//   1) q,k,v = x @ W{q,k,v}^T          -> V_WMMA_F32_16X16X4_F32 tiles
//   2) 3*16384 8-qubit circuits        -> 1 wave32 per circuit, state in VGPRs,
//                                         shfl_xor butterflies for lane-strides
//   3) degenerate attention softmax    -> block reduce over s=512
//   4) out = av @ Wo^T + bo            -> f32 WMMA again

typedef __attribute__((ext_vector_type(2))) float v2f;
typedef __attribute__((ext_vector_type(8))) float v8f;

#define EMBED 64
#define SEQ   512
#define BATCH 4
#define HEADS 8
#define ROWS  (BATCH * SEQ)          // 2048
#define NCIRC (ROWS * HEADS)         // 16384 circuits per matrix

// ---------------------------------------------------------------------------
// GEMM: out[m, j] = sum_k A[m,k] * W[j,k]  (+ optional bias[j])
// One wave computes one 16x16 output tile; blockDim=128 (4 waves = 4 N-tiles),
// gridDim.x = ROWS/16. K=64 consumed as 16 steps of V_WMMA_F32_16X16X4_F32.
// f32 WMMA lane layout (ISA 7.12.2):
//   A 16x4 : lanes 0-15 -> M=lane, K={0,1} in v0,v1 ; lanes 16-31 -> K={2,3}
//   B 4x16 : lanes 0-15 -> N=lane, K={0,1}          ; lanes 16-31 -> K={2,3}
//   C/D    : VGPR r, lanes 0-15 -> row r, col lane  ; lanes 16-31 -> row r+8
// ---------------------------------------------------------------------------
__global__ __launch_bounds__(128) void wmma_gemm_nt(
    const float* __restrict__ A,      // ROWS x 64, row-major
    const float* __restrict__ W,      // 64 x 64, row-major (used transposed)
    const float* __restrict__ bias,   // 64 (or null)
    float* __restrict__ out,          // ROWS x 64
    int use_bias)
{
  const int wave = threadIdx.x >> 5;
  const int lane = threadIdx.x & 31;
  const int m0 = blockIdx.x * 16;
  const int n0 = wave * 16;
  const int mrow = (lane < 16) ? lane : (lane - 16);
  const int koff = (lane < 16) ? 0 : 2;

  const float* arow = A + (m0 + mrow) * EMBED + koff;
  const float* brow = W + (n0 + mrow) * EMBED + koff;  // B[k][n] = W[n][k]

  v8f acc = {};
#pragma unroll
  for (int kk = 0; kk < 16; ++kk) {
    v2f a, b;
    a.x = arow[kk * 4 + 0];
    a.y = arow[kk * 4 + 1];
    b.x = brow[kk * 4 + 0];
    b.y = brow[kk * 4 + 1];
    acc = __builtin_amdgcn_wmma_f32_16x16x4_f32(
        /*neg_a=*/false, a, /*neg_b=*/false, b,
        /*c_mod=*/(short)0, acc, /*reuse_a=*/false, /*reuse_b=*/false);
  }

  const int col = n0 + mrow;
  const int rbase = (lane < 16) ? 0 : 8;
  const float bv = use_bias ? bias[col] : 0.0f;
#pragma unroll
  for (int r = 0; r < 8; ++r)
    out[(m0 + rbase + r) * EMBED + col] = acc[r] + bv;
}

// ---------------------------------------------------------------------------
// Quantum circuit: 8 qubits, 256 complex amps = 8 complex per lane of a wave32.
// Amplitude index idx = (v << 5) | lane  (v = register slot 0..7).
// Wire w pairs indices at stride d = 1 << (7-w):
//   d >= 32 -> register-pair butterfly within a lane
//   d <  32 -> cross-lane via __shfl_xor (wave32)
// RX butterfly is symmetric: n = c*self - i*s*partner.
// ---------------------------------------------------------------------------
__device__ __forceinline__ void rx_gate(float re[8], float im[8], float theta,
                                        int w, int lane)
{
  const float c = __cosf(theta * 0.5f);
  const float s = __sinf(theta * 0.5f);
  const int d = 1 << (7 - w);
  if (d >= 32) {
    const int dv = d >> 5;
#pragma unroll
    for (int v = 0; v < 8; ++v) {
      if ((v & dv) == 0) {
        const int u = v | dv;
        const float n0r = c * re[v] + s * im[u];
        const float n0i = c * im[v] - s * re[u];
        const float n1r = c * re[u] + s * im[v];
        const float n1i = c * im[u] - s * re[v];
        re[v] = n0r; im[v] = n0i;
        re[u] = n1r; im[u] = n1i;
      }
    }
  } else {
#pragma unroll
    for (int v = 0; v < 8; ++v) {
      const float pr = __shfl_xor(re[v], d, 32);
      const float pi = __shfl_xor(im[v], d, 32);
      const float nr = c * re[v] + s * pi;
      const float ni = c * im[v] - s * pr;
      re[v] = nr; im[v] = ni;
    }
  }
}

// CNOT(ctrl c, tgt t): where ctrl bit set, amplitude takes value from idx^tm.
__device__ __forceinline__ void cnot_gate(float re[8], float im[8], int c,
                                          int t, int lane)
{
  const int cm = 1 << (7 - c);
  const int tm = 1 << (7 - t);
  if (tm >= 32) {
    const int tv = tm >> 5;
    float orr[8], ori[8];
#pragma unroll
    for (int v = 0; v < 8; ++v) { orr[v] = re[v]; ori[v] = im[v]; }
#pragma unroll
    for (int v = 0; v < 8; ++v) {
      const int u = v ^ tv;
      const bool ctrl = (cm >= 32) ? ((v & (cm >> 5)) != 0)
                                   : ((lane & cm) != 0);
      re[v] = ctrl ? orr[u] : orr[v];
      im[v] = ctrl ? ori[u] : ori[v];
    }
  } else {
#pragma unroll
    for (int v = 0; v < 8; ++v) {
      const float pr = __shfl_xor(re[v], tm, 32);
      const float pi = __shfl_xor(im[v], tm, 32);
      const bool ctrl = (cm >= 32) ? ((v & (cm >> 5)) != 0)
                                   : ((lane & cm) != 0);
      re[v] = ctrl ? pr : re[v];
      im[v] = ctrl ? pi : im[v];
    }
  }
}

__global__ __launch_bounds__(256) void quantum_kernel(
    float* __restrict__ qbuf, float* __restrict__ kbuf,
    float* __restrict__ vbuf, const float* __restrict__ qparams)
{
  const int gwave = (int)((blockIdx.x * blockDim.x + threadIdx.x) >> 5);
  const int lane = threadIdx.x & 31;
  const int mat = gwave / NCIRC;        // 0=q, 1=k, 2=v
  const int cid = gwave % NCIRC;        // (b*SEQ+s)*HEADS + h
  const int h = cid & (HEADS - 1);
  const int row = cid >> 3;

  float* buf = (mat == 0) ? qbuf : (mat == 1) ? kbuf : vbuf;
  float* slot = buf + row * EMBED + h * 8;      // 8 angles in, 8 EVs out
  const float* lp = qparams + h * 8;            // (h, L=1, n=8)

  float re[8], im[8];
#pragma unroll
  for (int v = 0; v < 8; ++v) { re[v] = 0.0f; im[v] = 0.0f; }
  if (lane == 0) re[0] = 1.0f;                  // |0...0>

  // Encoding RX layer, then trainable RX layer.
#pragma unroll
  for (int w = 0; w < 8; ++w) rx_gate(re, im, slot[w], w, lane);
#pragma unroll
  for (int w = 0; w < 8; ++w) rx_gate(re, im, lp[w], w, lane);
  // Entangling CNOT ring: (0,1)...(6,7),(7,0).
#pragma unroll
  for (int c = 0; c < 8; ++c) cnot_gate(re, im, c, (c + 1) & 7, lane);

  // PauliZ expectation per wire: sum_p +/- |amp|^2.
  float ev[8];
#pragma unroll
  for (int i = 0; i < 8; ++i) ev[i] = 0.0f;
#pragma unroll
  for (int v = 0; v < 8; ++v) {
    const float p = re[v] * re[v] + im[v] * im[v];
    const int idx = (v << 5) | lane;
#pragma unroll
    for (int i = 0; i < 8; ++i)
      ev[i] += (idx & (1 << (7 - i))) ? -p : p;
  }
#pragma unroll
  for (int i = 0; i < 8; ++i) {
    float s = ev[i];
#pragma unroll
    for (int d = 16; d >= 1; d >>= 1) s += __shfl_xor(s, d, 32);
    ev[i] = s;
  }
  if (lane == 0) {
#pragma unroll
    for (int i = 0; i < 8; ++i) slot[i] = ev[i];
  }
}

// ---------------------------------------------------------------------------
// Attention (degenerate): score[b,h,s] = q[b,s,h,:].k[b,s,h,:]/sqrt(8);
// softmax over s; av[b,s,h*8+d] = attn[b,h,s] * v[b,s,h,d].
// One block per (b,h), 512 threads = seq positions.
// ---------------------------------------------------------------------------
__global__ __launch_bounds__(512) void attn_scale(
    const float* __restrict__ q, const float* __restrict__ k,
    const float* __restrict__ v, float* __restrict__ av)
{
  const int bh = blockIdx.x;
  const int b = bh >> 3, h = bh & 7;
  const int s = threadIdx.x;
  const int row = b * SEQ + s;
  const int lane = threadIdx.x & 31;
  const int wv = threadIdx.x >> 5;          // 16 waves

  const float* qp = q + row * EMBED + h * 8;
  const float* kp = k + row * EMBED + h * 8;
  float sc = 0.0f;
#pragma unroll
  for (int d = 0; d < 8; ++d) sc += qp[d] * kp[d];
  sc *= 0.35355339059327373f;               // 1/sqrt(8)

  __shared__ float smax[16];
  __shared__ float ssum[16];

  float m = sc;
#pragma unroll
  for (int d = 16; d >= 1; d >>= 1) m = fmaxf(m, __shfl_xor(m, d, 32));
  if (lane == 0) smax[wv] = m;
  __syncthreads();
  float gm = smax[0];
#pragma unroll
  for (int i = 1; i < 16; ++i) gm = fmaxf(gm, smax[i]);

  const float e = __expf(sc - gm);
  float t = e;
#pragma unroll
  for (int d = 16; d >= 1; d >>= 1) t += __shfl_xor(t, d, 32);
  if (lane == 0) ssum[wv] = t;
  __syncthreads();
  float gs = 0.0f;
#pragma unroll
  for (int i = 0; i < 16; ++i) gs += ssum[i];

  const float wgt = e / gs;
  const float* vp = v + row * EMBED + h * 8;
  float* op = av + row * EMBED + h * 8;
#pragma unroll
  for (int d = 0; d < 8; ++d) op[d] = wgt * vp[d];
}

// ---------------------------------------------------------------------------
extern "C" void kernel_launch(void* const* d_in, const int* in_sizes, int n_in,
                              void* d_out, int out_size, void* d_ws,
                              size_t ws_size, hipStream_t stream)
{
  const float* x       = (const float*)d_in[0];
  const float* Wq      = (const float*)d_in[1];
  const float* Wk      = (const float*)d_in[2];
  const float* Wv      = (const float*)d_in[3];
  const float* Wo      = (const float*)d_in[4];
  const float* bo      = (const float*)d_in[5];
  const float* qparams = (const float*)d_in[6];
  float* out = (float*)d_out;

  // Workspace: q, k, v, av -> 4 * 2048*64*4B = 2 MB.
  float* q  = (float*)d_ws;
  float* k  = q + ROWS * EMBED;
  float* v  = k + ROWS * EMBED;
  float* av = v + ROWS * EMBED;

  const dim3 gemm_grid(ROWS / 16);
  const dim3 gemm_block(128);
  wmma_gemm_nt<<<gemm_grid, gemm_block, 0, stream>>>(x, Wq, nullptr, q, 0);
  wmma_gemm_nt<<<gemm_grid, gemm_block, 0, stream>>>(x, Wk, nullptr, k, 0);
  wmma_gemm_nt<<<gemm_grid, gemm_block, 0, stream>>>(x, Wv, nullptr, v, 0);

  // 3 * 16384 circuits, one wave32 each; 8 waves per 256-thread block.
  const int total_waves = 3 * NCIRC;
  quantum_kernel<<<total_waves / 8, 256, 0, stream>>>(q, k, v, qparams);

  attn_scale<<<BATCH * HEADS, SEQ, 0, stream>>>(q, k, v, av);

  wmma_gemm_nt<<<gemm_grid, gemm_block, 0, stream>>>(av, Wo, bo, out, 1);
}